// MPS_81612968559318
// MI455X (gfx1250) — compile-verified
//
#include <hip/hip_runtime.h>

// MPS chain contraction for MI455X (gfx1250, wave32, WMMA).
//
// Per site s, per batch b:  M[b,s] = x0*C0[s] + x1*C1[s]  (64x64).
// Left half needs  e0^T * prod M,  right half needs  prod M * e0.
// Batches stacked as columns of a 64-wide state matrix turn one chain step into
// two shared 64x64 fp32 GEMMs + a per-lane (per-batch) scalar combine:
//   right: U <- x0.*(C0 @ U)   + x1.*(C1 @ U)
//   left : V <- x0.*(C0^T @ V) + x1.*(C1^T @ V)
// V_WMMA_F32_16X16X4_F32 keeps fp32 precision across the 98-deep product.
//
// Roofline: 0.21 GFLOP / 6.7 MB -> ~0.3us memory floor; the bottleneck is the
// 98-step serial dependency. Design: 8 WGs (4 batch-groups x {L,R}), 4 waves,
// one 16x16 tile per wave -> 32 back-to-back WMMAs per site per wave;
// double-buffered LDS cores with register staging pipelines the 32KB/site
// global traffic two sites ahead; single barrier per site.

typedef float v2f __attribute__((ext_vector_type(2)));
typedef float v8f __attribute__((ext_vector_type(8)));

#define NSITES   196
#define HALF     98
#define DBOND    64
#define NB       16      // batches per workgroup
#define THREADS  128     // 4 waves (wave32)
#define SITE_F4  512     // float4 stride between sites (a site is 2*64*64 floats = 2048 float4)

__global__ __launch_bounds__(THREADS)
void mps_chain_kernel(const float* __restrict__ xg,       // (64,196,2)
                      const float* __restrict__ coresG,   // (196,64,64,2) C0/C1 interleaved
                      float* __restrict__ wsV,            // (64,64) [bond][batch]
                      float* __restrict__ wsU)            // (64,64) [bond][batch]
{
    const int  blk  = blockIdx.x;
    const bool left = (blk < 4);
    const int  bb   = (blk & 3) * NB;        // batch-group base
    const int  sBeg = left ? 0 : HALF;

    const int tid  = threadIdx.x;
    const int lane = tid & 31;
    const int wave = tid >> 5;
    const int half = lane >> 4;              // lane-half (WMMA layout)
    const int n    = lane & 15;              // N index = local batch
    const int m0   = wave * 16;              // M-tile base (one tile per wave)

    __shared__ float ldsC[2][2 * DBOND * DBOND]; // double-buffered site cores [r][c][i]
    __shared__ float Sst[2][DBOND * NB];         // ping-pong state [k][n]
    __shared__ float xls[NB * 2 * HALF];         // x slice for this WG

    // Preload x (contiguous per batch: (s,i) are the fastest dims).
    for (int t = tid; t < NB * 2 * HALF; t += THREADS) {
        int nb = t / (2 * HALF), j = t % (2 * HALF);
        xls[t] = xg[(bb + nb) * (2 * NSITES) + sBeg * 2 + j];
    }
    // State init: every batch column = e0 (k==0 <=> t<16).
    for (int t = tid; t < DBOND * NB; t += THREADS)
        Sst[0][t] = (t < NB) ? 1.0f : 0.0f;

    // Site address helper: left walks 0..97, right walks 97..0 (M195 applied first).
    const float4* gC = reinterpret_cast<const float4*>(coresG);
    auto siteF4 = [&](int st) -> const float4* {
        const int sl = left ? st : (HALF - 1 - st);
        return gC + (size_t)(sBeg + sl) * SITE_F4 + tid;
    };

    // Pipeline prologue: site0 -> LDS buffer 0; site1 -> staging regs.
    float4 stg[16];
    {
        const float4* g0 = siteF4(0);
        #pragma unroll
        for (int i = 0; i < 16; ++i) stg[i] = g0[i * THREADS];
        float4* l0 = reinterpret_cast<float4*>(ldsC[0]) + tid;
        #pragma unroll
        for (int i = 0; i < 16; ++i) l0[i * THREADS] = stg[i];
        const float4* g1 = siteF4(1);
        #pragma unroll
        for (int i = 0; i < 16; ++i) stg[i] = g1[i * THREADS];
    }
    __syncthreads();

    int cur = 0;
    for (int step = 0; step < HALF; ++step) {
        const int cb = step & 1;
        const int sl = left ? step : (HALF - 1 - step);

        // Drain staging regs (site step+1) into the spare LDS buffer, then
        // issue global loads for site step+2 (a full WMMA chain to cover them).
        if (step + 1 < HALF) {
            float4* ld = reinterpret_cast<float4*>(ldsC[cb ^ 1]) + tid;
            #pragma unroll
            for (int i = 0; i < 16; ++i) ld[i * THREADS] = stg[i];
        }
        if (step + 2 < HALF) {
            const float4* gn = siteF4(step + 2);
            #pragma unroll
            for (int i = 0; i < 16; ++i) stg[i] = gn[i * THREADS];
        }

        const float* C = ldsC[cb];

        // B fragments (state), identical for all waves.
        // B 4x16 layout: v0 = rows k0 | k0+2 (lane halves), v1 = rows k0+1 | k0+3.
        v2f bf[16];
        {
            const float* sp   = Sst[cur];
            const int    base = (2 * half) * NB + n;
            #pragma unroll
            for (int j = 0; j < 16; ++j) {
                bf[j][0] = sp[j * (4 * NB) + base];
                bf[j][1] = sp[j * (4 * NB) + base + NB];
            }
        }

        // Preload ALL A fragments, then run 32 WMMAs back-to-back (no dscnt
        // waits inside the matrix chain). A 16x4 fp32 layout: lanes<16 hold
        // K={k0,k0+1}, lanes>=16 hold K={k0+2,k0+3}.
        v2f a0[16], a1[16];
        if (left) {
            #pragma unroll
            for (int j = 0; j < 16; ++j) {   // A = C^T: A[m,k] = C[k][m]
                const int idx = ((4 * j + 2 * half) * DBOND + m0 + n) * 2;
                a0[j][0] = C[idx];       a1[j][0] = C[idx + 1];
                a0[j][1] = C[idx + 128]; a1[j][1] = C[idx + 129];
            }
        } else {
            #pragma unroll
            for (int j = 0; j < 16; ++j) {   // A = C: 16B read -> (C0,C1)x(k,k+1)
                const int idx = ((m0 + n) * DBOND + 4 * j + 2 * half) * 2;
                const float4 q = *reinterpret_cast<const float4*>(&C[idx]);
                a0[j][0] = q.x; a1[j][0] = q.y; a0[j][1] = q.z; a1[j][1] = q.w;
            }
        }

        v8f acc0 = {0.f, 0.f, 0.f, 0.f, 0.f, 0.f, 0.f, 0.f};
        v8f acc1 = acc0;
        #pragma unroll
        for (int j = 0; j < 16; ++j) {
            acc0 = __builtin_amdgcn_wmma_f32_16x16x4_f32(false, a0[j], false, bf[j],
                                                         (short)0, acc0, false, false);
            acc1 = __builtin_amdgcn_wmma_f32_16x16x4_f32(false, a1[j], false, bf[j],
                                                         (short)0, acc1, false, false);
        }

        // Per-batch scalar combine; x depends only on N (= lane).
        const float x0 = xls[n * (2 * HALF) + sl * 2 + 0];
        const float x1 = xls[n * (2 * HALF) + sl * 2 + 1];
        float* snew = Sst[cur ^ 1];
        #pragma unroll
        for (int g = 0; g < 8; ++g) {               // D layout: M = m0 + g + 8*half
            const int m = m0 + g + 8 * half;
            snew[m * NB + n] = x0 * acc0[g] + x1 * acc1[g];
        }

        // Single barrier per site: publishes new state AND the cores buffer
        // for site step+1 (its reads in this iteration were consumed above).
        __syncthreads();
        cur ^= 1;
    }

    // Publish final state columns to workspace.
    float* dst = left ? wsV : wsU;
    for (int t = tid; t < DBOND * NB; t += THREADS) {
        const int k = t >> 4, nn = t & 15;
        dst[k * 64 + bb + nn] = Sst[cur][t];
    }
}

// out[b,o] = sum_{l,r} V[l,b] * W[o,l,r] * U[r,b]   (~5 MFLOP, trivial)
__global__ __launch_bounds__(64)
void mps_out_kernel(const float* __restrict__ wsV, const float* __restrict__ wsU,
                    const float* __restrict__ W, float* __restrict__ out)
{
    const int o = blockIdx.x;   // 0..9
    const int b = threadIdx.x;  // 0..63
    float u[64];
    #pragma unroll
    for (int r = 0; r < 64; ++r) u[r] = wsU[r * 64 + b];
    float acc = 0.f;
    for (int l = 0; l < 64; ++l) {
        const float  vl = wsV[l * 64 + b];
        const float* wr = W + (o * 64 + l) * 64;
        float inner = 0.f;
        #pragma unroll
        for (int r = 0; r < 64; ++r) inner += wr[r] * u[r];
        acc += vl * inner;
    }
    out[b * 10 + o] = acc;   // output shape (B, O)
}

extern "C" void kernel_launch(void* const* d_in, const int* in_sizes, int n_in,
                              void* d_out, int out_size, void* d_ws, size_t ws_size,
                              hipStream_t stream)
{
    const float* x        = (const float*)d_in[0];  // (64,196,2)
    const float* cores    = (const float*)d_in[1];  // (196,64,64,2)
    const float* out_core = (const float*)d_in[2];  // (10,64,64)
    float* out = (float*)d_out;                     // (64,10)
    float* wsV = (float*)d_ws;                      // 4096 floats
    float* wsU = wsV + 64 * 64;                     // 4096 floats

    hipLaunchKernelGGL(mps_chain_kernel, dim3(8), dim3(THREADS), 0, stream,
                       x, cores, wsV, wsU);
    hipLaunchKernelGGL(mps_out_kernel, dim3(10), dim3(64), 0, stream,
                       wsV, wsU, out_core, out);
}